// Net_16767552324115
// MI455X (gfx1250) — compile-verified
//
#include <hip/hip_runtime.h>
#include <hip/hip_bf16.h>

typedef __attribute__((ext_vector_type(2))) float v2f;
typedef __attribute__((ext_vector_type(8))) float v8f;

#define FIN 512
#define F1  16
#define F2  7

// ---------- degree / normalization ----------
__global__ void k_deg_init(float* deg, int n) {
  int i = blockIdx.x * blockDim.x + threadIdx.x;
  if (i < n) deg[i] = 1.0f;  // self-loop contribution
}

__global__ void k_deg_count(const int* __restrict__ dst, float* deg, int e) {
  int i = blockIdx.x * blockDim.x + threadIdx.x;
  if (i < e) atomicAdd(&deg[dst[i]], 1.0f);
}

__global__ void k_rsqrt(const float* __restrict__ deg, float* dis, int n) {
  int i = blockIdx.x * blockDim.x + threadIdx.x;
  if (i < n) dis[i] = rsqrtf(deg[i]);
}

// ---------- h1[N,16] = x[N,512] @ W1[512,16] via V_WMMA_F32_16X16X4_F32 ----------
// One wave handles a 16-row tile of x. A: 16x4 f32 (2 VGPR), B: 4x16 f32 (2 VGPR),
// C/D: 16x16 f32 (8 VGPR). K loop: 512/4 = 128 WMMAs per tile.
__global__ void k_gemm1_wmma(const float* __restrict__ x, const float* __restrict__ W1,
                             float* __restrict__ h1, int n) {
  int wave = (int)((blockIdx.x * (int)blockDim.x + (int)threadIdx.x) >> 5);
  int lane = (int)threadIdx.x & 31;
  int row0 = wave * 16;
  if (row0 >= n) return;  // wave-uniform: EXEC stays all-1s for active waves
  int m    = lane & 15;   // A: matrix row   / B: matrix column
  int half = lane >> 4;   // 0: K={0,1}, 1: K={2,3} within each 4-K slab
  int kp   = half * 2;
  const float* xr = x  + (size_t)(row0 + m) * FIN + kp;   // A source (row-major x)
  const float* wc = W1 + (size_t)kp * F1 + m;             // B source (row-major W1)
  v8f c = {};
#pragma unroll 8
  for (int k0 = 0; k0 < FIN; k0 += 4) {
    v2f a, b;
    a.x = xr[k0];
    a.y = xr[k0 + 1];
    b.x = wc[(size_t)k0 * F1];
    b.y = wc[(size_t)(k0 + 1) * F1];
    c = __builtin_amdgcn_wmma_f32_16x16x4_f32(false, a, false, b, (short)0, c,
                                              false, false);
  }
  // D layout: VGPR r -> row (row0 + r + half*8), col m
  float* o = h1 + (size_t)(row0 + half * 8) * F1 + m;
#pragma unroll
  for (int r = 0; r < 8; ++r) o[(size_t)r * F1] = c[r];
}

// ---------- layer-1 aggregation ----------
__global__ void k_self1(const float* __restrict__ h1, const float* __restrict__ dis,
                        float* out1, int n) {
  int j = blockIdx.x * blockDim.x + threadIdx.x;
  if (j < n * F1) {
    float d = dis[j >> 4];
    out1[j] = h1[j] * d * d;  // self-loop term
  }
}

__global__ void k_agg1(const int* __restrict__ src, const int* __restrict__ dst,
                       const float* __restrict__ dis, const float* __restrict__ h1,
                       float* out1, int e) {
  int i = blockIdx.x * blockDim.x + threadIdx.x;
  if (i >= e) return;
  int s = src[i], d = dst[i];
  float w = dis[s] * dis[d];
  const float4* hs = (const float4*)(h1 + (size_t)s * F1);
  float* od = out1 + (size_t)d * F1;
#pragma unroll
  for (int q = 0; q < 4; ++q) {
    float4 v = hs[q];
    atomicAdd(od + q * 4 + 0, v.x * w);
    atomicAdd(od + q * 4 + 1, v.y * w);
    atomicAdd(od + q * 4 + 2, v.z * w);
    atomicAdd(od + q * 4 + 3, v.w * w);
  }
}

__global__ void k_bias_relu(const float* __restrict__ out1, const float* __restrict__ b1,
                            float* hr, int n) {
  int j = blockIdx.x * blockDim.x + threadIdx.x;
  if (j < n * F1) {
    float v = out1[j] + b1[j & 15];
    hr[j] = v > 0.f ? v : 0.f;
  }
}

// ---------- h2[N,7] = hr[N,16] @ W2[16,7] ----------
__global__ void k_gemm2(const float* __restrict__ hr, const float* __restrict__ W2,
                        float* h2, int n) {
  int i = blockIdx.x * blockDim.x + threadIdx.x;
  if (i >= n) return;
  const float* h = hr + (size_t)i * F1;
  float acc[F2];
#pragma unroll
  for (int c = 0; c < F2; ++c) acc[c] = 0.f;
#pragma unroll
  for (int k = 0; k < F1; ++k) {
    float hv = h[k];
#pragma unroll
    for (int c = 0; c < F2; ++c) acc[c] += hv * W2[k * F2 + c];
  }
  float* o = h2 + (size_t)i * F2;
#pragma unroll
  for (int c = 0; c < F2; ++c) o[c] = acc[c];
}

// ---------- layer-2 aggregation (accumulate directly into d_out) ----------
__global__ void k_self2(const float* __restrict__ h2, const float* __restrict__ dis,
                        float* out2, int n) {
  int j = blockIdx.x * blockDim.x + threadIdx.x;
  if (j < n * F2) {
    float d = dis[j / F2];
    out2[j] = h2[j] * d * d;
  }
}

__global__ void k_agg2(const int* __restrict__ src, const int* __restrict__ dst,
                       const float* __restrict__ dis, const float* __restrict__ h2,
                       float* out2, int e) {
  int i = blockIdx.x * blockDim.x + threadIdx.x;
  if (i >= e) return;
  int s = src[i], d = dst[i];
  float w = dis[s] * dis[d];
  const float* hs = h2 + (size_t)s * F2;
  float* od = out2 + (size_t)d * F2;
#pragma unroll
  for (int c = 0; c < F2; ++c) atomicAdd(od + c, hs[c] * w);
}

__global__ void k_logsoftmax(float* out2, const float* __restrict__ b2, int n) {
  int i = blockIdx.x * blockDim.x + threadIdx.x;
  if (i >= n) return;
  float* p = out2 + (size_t)i * F2;
  float v[F2];
  v[0] = p[0] + b2[0];
  float m = v[0];
#pragma unroll
  for (int c = 1; c < F2; ++c) {
    v[c] = p[c] + b2[c];
    m = fmaxf(m, v[c]);
  }
  float s = 0.f;
#pragma unroll
  for (int c = 0; c < F2; ++c) s += __expf(v[c] - m);
  float l = __logf(s);
#pragma unroll
  for (int c = 0; c < F2; ++c) p[c] = v[c] - m - l;
}

extern "C" void kernel_launch(void* const* d_in, const int* in_sizes, int n_in,
                              void* d_out, int out_size, void* d_ws, size_t ws_size,
                              hipStream_t stream) {
  const float* x  = (const float*)d_in[0];
  const int*   ei = (const int*)d_in[1];
  const float* W1 = (const float*)d_in[2];
  const float* b1 = (const float*)d_in[3];
  const float* W2 = (const float*)d_in[4];
  const float* b2 = (const float*)d_in[5];

  int n = in_sizes[0] / FIN;  // 200000
  int e = in_sizes[1] / 2;    // 6400000
  const int* src = ei;
  const int* dst = ei + e;

  float* deg  = (float*)d_ws;                 // N
  float* dis  = deg  + n;                     // N
  float* h1   = dis  + n;                     // 16N (reused as hr)
  float* out1 = h1   + (size_t)n * F1;        // 16N
  float* h2   = out1 + (size_t)n * F1;        // 7N
  float* out2 = (float*)d_out;                // 7N (final output buffer)

  const int B = 256;
  auto cdiv = [](long long a, long long b) { return (int)((a + b - 1) / b); };

  k_deg_init  <<<cdiv(n, B), B, 0, stream>>>(deg, n);
  k_deg_count <<<cdiv(e, B), B, 0, stream>>>(dst, deg, e);
  k_rsqrt     <<<cdiv(n, B), B, 0, stream>>>(deg, dis, n);

  int tiles = cdiv(n, 16);  // one wave (32 lanes) per 16-row tile
  k_gemm1_wmma<<<cdiv((long long)tiles * 32, B), B, 0, stream>>>(x, W1, h1, n);

  k_self1     <<<cdiv((long long)n * F1, B), B, 0, stream>>>(h1, dis, out1, n);
  k_agg1      <<<cdiv(e, B), B, 0, stream>>>(src, dst, dis, h1, out1, e);
  k_bias_relu <<<cdiv((long long)n * F1, B), B, 0, stream>>>(out1, b1, h1, n);
  k_gemm2     <<<cdiv(n, B), B, 0, stream>>>(h1, W2, h2, n);
  k_self2     <<<cdiv((long long)n * F2, B), B, 0, stream>>>(h2, dis, out2, n);
  k_agg2      <<<cdiv(e, B), B, 0, stream>>>(src, dst, dis, h2, out2, e);
  k_logsoftmax<<<cdiv(n, B), B, 0, stream>>>(out2, b2, n);
}